// ContentAlignment_20641612824793
// MI455X (gfx1250) — compile-verified
//
#include <hip/hip_runtime.h>
#include <hip/hip_bf16.h>
#include <stdint.h>

// ---- static shapes from the reference setup ----
#define BATCH   128
#define IMG_H   128
#define IMG_W   1024
#define OUT_H   128
#define OUT_W   1024
#define TEXT_L  64
#define CHAR_H  128
#define CHAR_W  16
#define PAD_VAL (-1.0f)

typedef float v4f __attribute__((ext_vector_type(4)));   // native clang vector (b128)

// params layout per sample: [th, tw, mh, mw]

// ---------------------------------------------------------------------------
// Kernel 1: text lengths. One 64-thread block per sample.
// ---------------------------------------------------------------------------
__global__ __launch_bounds__(64)
void text_reduce_kernel(const int* __restrict__ text, int* __restrict__ params) {
    const int b = blockIdx.x;
    const int t = threadIdx.x;
    __shared__ int ssum[64];
    __shared__ int scnt[64];
    const int v = text[b * TEXT_L + t];
    ssum[t] = v;
    scnt[t] = (v != 0) ? 1 : 0;
    __syncthreads();
    #pragma unroll
    for (int s = 32; s > 0; s >>= 1) {
        if (t < s) { ssum[t] += ssum[t + s]; scnt[t] += scnt[t + s]; }
        __syncthreads();
    }
    if (t == 0) {
        int th = (ssum[0] != 0) ? ((CHAR_H < IMG_H) ? CHAR_H : IMG_H) : 0;
        int tw = scnt[0] * CHAR_W;
        tw = (tw < 0) ? 0 : ((tw > IMG_W) ? IMG_W : tw);
        params[b * 4 + 0] = th;
        params[b * 4 + 1] = tw;
    }
}

// ---------------------------------------------------------------------------
// Kernel 2: mask content extents. One 256-thread block per sample.
// Rows (4 KB each) stream global->LDS via the CDNA5 async data path
// (global_load_async_to_lds_b128, ASYNCcnt), double-buffered: row r+1's
// fetch overlaps row r's nonzero test.
// ---------------------------------------------------------------------------
__global__ __launch_bounds__(256)
void mask_reduce_kernel(const float* __restrict__ mask, int* __restrict__ params) {
    const int b = blockIdx.x;
    const int t = threadIdx.x;                 // 0..255, owns columns 4t..4t+3

    __shared__ __align__(16) float rowbuf[2][OUT_W];
    __shared__ int rowf[OUT_H];
    __shared__ int cnts[2];                    // [0]=mh, [1]=mw

    if (t < OUT_H) rowf[t] = 0;
    if (t < 2)     cnts[t] = 0;
    __syncthreads();

    const unsigned long long gb =
        (unsigned long long)(uintptr_t)(mask + (size_t)b * (OUT_H * OUT_W));
    const uint32_t lds0 = (uint32_t)(uintptr_t)(&rowbuf[0][t * 4]);
    const uint32_t lds1 = (uint32_t)(uintptr_t)(&rowbuf[1][t * 4]);
    const uint32_t lane_byte = (uint32_t)(t * 16);

    // issue row 0
    {
        uint32_t voff = lane_byte;             // r=0
        asm volatile("global_load_async_to_lds_b128 %0, %1, %2"
                     :: "v"(lds0), "v"(voff), "s"(gb) : "memory");
    }

    int c0 = 0, c1 = 0, c2 = 0, c3 = 0;        // column nonzero flags (registers)

    for (int r = 0; r < OUT_H; ++r) {
        if (r + 1 < OUT_H) {
            uint32_t voff = (uint32_t)((r + 1) * (OUT_W * 4)) + lane_byte;
            uint32_t ldsn = ((r + 1) & 1) ? lds1 : lds0;
            asm volatile("global_load_async_to_lds_b128 %0, %1, %2"
                         :: "v"(ldsn), "v"(voff), "s"(gb) : "memory");
            asm volatile("s_wait_asynccnt 0x1" ::: "memory");
        } else {
            asm volatile("s_wait_asynccnt 0x0" ::: "memory");
        }
        const v4f v = *(const v4f*)(&rowbuf[r & 1][t * 4]);
        const int a0 = (v.x != 0.0f), a1 = (v.y != 0.0f);
        const int a2 = (v.z != 0.0f), a3 = (v.w != 0.0f);
        c0 |= a0; c1 |= a1; c2 |= a2; c3 |= a3;
        if (a0 | a1 | a2 | a3) rowf[r] = 1;    // benign race: all writers store 1
    }
    __syncthreads();

    atomicAdd(&cnts[1], c0 + c1 + c2 + c3);
    if (t < OUT_H) atomicAdd(&cnts[0], rowf[t]);
    __syncthreads();

    if (t == 0) {
        params[b * 4 + 2] = cnts[0];           // mh
        params[b * 4 + 3] = cnts[1];           // mw
    }
}

// ---------------------------------------------------------------------------
// Kernel 3: content alignment + mask. 4 consecutive pixels per thread:
// b128 mask load + b128 non-temporal store; vertical resample math hoisted
// (row-invariant across the quad). 128 blocks per sample -> uniform params.
// ---------------------------------------------------------------------------
__global__ __launch_bounds__(256)
void align_kernel(const float* __restrict__ img,
                  const float* __restrict__ mask,
                  const int*   __restrict__ params,
                  float*       __restrict__ out) {
    const int bid  = blockIdx.x;
    const int b    = bid >> 7;                            // 128 blocks/sample
    const int quad = ((bid & 127) << 8) | threadIdx.x;    // 0..32767
    const int pix0 = quad << 2;                           // 4 consecutive pixels
    const int y    = pix0 >> 10;                          // OUT_W = 1024
    const int xb   = pix0 & (OUT_W - 1);                  // xb..xb+3 (same row)

    const int th = params[b * 4 + 0];
    const int tw = params[b * 4 + 1];
    const int mh = params[b * 4 + 2];
    const int mw = params[b * 4 + 3];

    const size_t sbase = (size_t)b * (size_t)(OUT_H * OUT_W);
    const v4f mv = *(const v4f*)(&mask[sbase + pix0]);

    const bool has_right  = (IMG_W > tw) && (OUT_W > mw);
    const bool has_bottom = (IMG_H > th) && (OUT_H > mh);
    const bool vb         = (y >= mh) && has_bottom;      // bottom band (all x)
    const bool row_top    = (y < mh);

    // ---- vertical resample terms (row-invariant across the quad) ----
    const int   oy  = vb ? th : 0;
    const int   sh  = vb ? (IMG_H - th) : th;
    const int   dh  = vb ? (OUT_H - mh) : mh;
    const int   dy0 = vb ? mh : 0;
    const float shf = (float)((sh > 1) ? sh : 1);
    const float dhf = (float)((dh > 1) ? dh : 1);
    float sy = ((float)(y - dy0) + 0.5f) * shf / dhf - 0.5f;
    sy = fminf(fmaxf(sy, 0.0f), shf - 1.0f);
    const float y0f = floorf(sy);
    const float fy  = sy - y0f;
    const int   y0i = (int)y0f;
    const int   shm1 = (sh - 1 > 0) ? (sh - 1) : 0;
    const int   y1i = (y0i + 1 < shm1) ? (y0i + 1) : shm1;
    int iy0 = oy + y0i; iy0 = (iy0 < 0) ? 0 : ((iy0 > IMG_H - 1) ? IMG_H - 1 : iy0);
    int iy1 = oy + y1i; iy1 = (iy1 < 0) ? 0 : ((iy1 > IMG_H - 1) ? IMG_H - 1 : iy1);

    const float* __restrict__ ib = img + (size_t)b * (size_t)(IMG_H * IMG_W);
    const float* __restrict__ r0 = ib + iy0 * IMG_W;
    const float* __restrict__ r1 = ib + iy1 * IMG_W;

    float res[4];
    const float mvals[4] = {mv.x, mv.y, mv.z, mv.w};

    #pragma unroll
    for (int k = 0; k < 4; ++k) {
        const int  x        = xb + k;
        const bool in_main  = row_top && (x < mw);
        const bool in_right = row_top && (x >= mw) && has_right;
        const bool is_pad   = !(in_main || in_right || vb);

        float val = PAD_VAL;
        if (!is_pad) {
            const int   ox  = in_right ? tw : 0;
            const int   sw  = in_right ? (IMG_W - tw) : tw;
            const int   dw  = in_right ? (OUT_W - mw) : (vb ? OUT_W : mw);
            const int   dx0 = in_right ? mw : 0;
            const float swf = (float)((sw > 1) ? sw : 1);
            const float dwf = (float)((dw > 1) ? dw : 1);

            float sx = ((float)(x - dx0) + 0.5f) * swf / dwf - 0.5f;
            sx = fminf(fmaxf(sx, 0.0f), swf - 1.0f);
            const float x0f = floorf(sx);
            const float fx  = sx - x0f;
            const int   x0i = (int)x0f;
            const int   swm1 = (sw - 1 > 0) ? (sw - 1) : 0;
            const int   x1i = (x0i + 1 < swm1) ? (x0i + 1) : swm1;
            int ix0 = ox + x0i; ix0 = (ix0 < 0) ? 0 : ((ix0 > IMG_W - 1) ? IMG_W - 1 : ix0);
            int ix1 = ox + x1i; ix1 = (ix1 < 0) ? 0 : ((ix1 > IMG_W - 1) ? IMG_W - 1 : ix1);

            const float v00 = r0[ix0];
            const float v01 = r0[ix1];
            const float v10 = r1[ix0];
            const float v11 = r1[ix1];

            val = (v00 * (1.0f - fy) + v10 * fy) * (1.0f - fx)
                + (v01 * (1.0f - fy) + v11 * fy) * fx;
        }
        const float m = mvals[k];
        res[k] = val * m + fminf(fmaxf(m - 1.0f, PAD_VAL), 0.0f);
    }

    v4f o;
    o.x = res[0]; o.y = res[1]; o.z = res[2]; o.w = res[3];
    __builtin_nontemporal_store(o, (v4f*)(&out[sbase + pix0]));  // write-once
}

// ---------------------------------------------------------------------------
extern "C" void kernel_launch(void* const* d_in, const int* in_sizes, int n_in,
                              void* d_out, int out_size, void* d_ws, size_t ws_size,
                              hipStream_t stream) {
    const float* img  = (const float*)d_in[0];   // [B,H,W,1] f32
    const int*   text = (const int*)d_in[1];     // [B,L] i32
    const float* mask = (const float*)d_in[2];   // [B,Ht,Wt,1] f32
    float*       out  = (float*)d_out;           // [B,Ht,Wt,1] f32
    int*         prm  = (int*)d_ws;              // [B,4] ints

    text_reduce_kernel<<<BATCH, 64, 0, stream>>>(text, prm);
    mask_reduce_kernel<<<BATCH, 256, 0, stream>>>(mask, prm);
    align_kernel<<<BATCH * 128, 256, 0, stream>>>(img, mask, prm, out);
}